// KNNSelector_52664888984289
// MI455X (gfx1250) — compile-verified
//
#include <hip/hip_runtime.h>
#include <hip/hip_bf16.h>

// KNN selector for MI455X (gfx1250, wave32).
//   q:[256,128] f32, k:[10000,128] f32, obs:[10000,4,84,84] f32
//   out:[10,256,4,84,84] f32
// Pipeline: k2 -> WMMA f32 distance tiles -> per-query wave top-10 -> big gather.

typedef float v2f __attribute__((ext_vector_type(2)));
typedef float v4f __attribute__((ext_vector_type(4)));
typedef float v8f __attribute__((ext_vector_type(8)));

#define QN   256
#define NN   10000
#define DD   128
#define KSEL 10
#define CHW  28224          // 4*84*84
#define CHW4 7056           // CHW/4

// ---------------------------------------------------------------------------
// k2[n] = sum_d k[n][d]^2
// ---------------------------------------------------------------------------
__global__ void knn_k2_kernel(const float* __restrict__ k, float* __restrict__ k2) {
    int n = blockIdx.x * blockDim.x + threadIdx.x;
    if (n >= NN) return;
    const v4f* row = (const v4f*)(k + (size_t)n * DD);
    float acc = 0.0f;
#pragma unroll
    for (int i = 0; i < DD / 4; ++i) {
        v4f v = row[i];
        acc += v.x * v.x + v.y * v.y + v.z * v.z + v.w * v.w;
    }
    k2[n] = acc;
}

// ---------------------------------------------------------------------------
// s[qrow][n] = k2[n] - 2 * dot(q[qrow], k[n])   (q^2 term dropped: constant
// per row, ordering-invariant for top-k). One wave32 per 16x16 tile, full-f32
// WMMA: V_WMMA_F32_16X16X4_F32, K-loop of 32 steps over D=128.
//
// A (16x4 f32) lane layout: lanes 0-15 hold M=lane, K={0,1} in v[0],v[1];
// lanes 16-31 hold M=lane-16, K={2,3}. B (4x16) mirrors it (K rows striped
// across VGPRs, N across lanes). Both fragments are the same float2 load
// pattern: row*128 + kk + 2*(lane>>4).
// ---------------------------------------------------------------------------
__global__ void knn_dist_wmma(const float* __restrict__ q,
                              const float* __restrict__ k,
                              const float* __restrict__ k2,
                              float* __restrict__ s) {
    const int nt   = blockIdx.x;          // 0..624  (N tile)
    const int qt   = blockIdx.y;          // 0..15   (Q tile)
    const int lane = threadIdx.x;         // 0..31
    const int half = lane >> 4;           // 0/1
    const int l16  = lane & 15;

    const float* qrow = q + (size_t)(qt * 16 + l16) * DD;
    const float* krow = k + (size_t)(nt * 16 + l16) * DD;

    v8f acc = {};
#pragma unroll
    for (int kk = 0; kk < DD; kk += 4) {
        const int o = kk + 2 * half;
        v2f a = *(const v2f*)(qrow + o);      // A[M=l16][K=o..o+1]
        v2f b = *(const v2f*)(krow + o);      // B[K=o..o+1][N=l16]
        acc = __builtin_amdgcn_wmma_f32_16x16x4_f32(
            /*neg_a=*/false, a, /*neg_b=*/false, b,
            /*c_mod=*/(short)0, acc, /*reuse_a=*/false, /*reuse_b=*/false);
    }

    // C/D layout: VGPR r -> M = r + 8*half, N = l16.
    const int col = nt * 16 + l16;
    const float k2v = k2[col];
#pragma unroll
    for (int r = 0; r < 8; ++r) {
        const int row = qt * 16 + r + 8 * half;
        s[(size_t)row * NN + col] = k2v - 2.0f * acc[r];
    }
}

// ---------------------------------------------------------------------------
// Per-query top-10 (smallest). One wave32 per query. Each lane keeps a sorted
// top-10 in registers over a strided scan, then 10 rounds of butterfly
// min-with-index reduction merge the 32 lists. Ties -> lower index (stable,
// matching jax.lax.top_k ordering on -dist).
// ---------------------------------------------------------------------------
__global__ void knn_topk(const float* __restrict__ s, int* __restrict__ idx) {
    const int qq   = blockIdx.x;     // 0..255
    const int lane = threadIdx.x;    // 0..31
    const float* row = s + (size_t)qq * NN;

    float bv[KSEL];
    int   bi[KSEL];
#pragma unroll
    for (int i = 0; i < KSEL; ++i) { bv[i] = 3.4e38f; bi[i] = 0x7fffffff; }

    for (int n = lane; n < NN; n += 32) {
        float v = row[n];
        if (v < bv[KSEL - 1] || (v == bv[KSEL - 1] && n < bi[KSEL - 1])) {
            bv[KSEL - 1] = v;
            bi[KSEL - 1] = n;
#pragma unroll
            for (int j = KSEL - 1; j > 0; --j) {
                bool swp = (bv[j] < bv[j - 1]) ||
                           (bv[j] == bv[j - 1] && bi[j] < bi[j - 1]);
                if (swp) {
                    float tv = bv[j]; bv[j] = bv[j - 1]; bv[j - 1] = tv;
                    int   ti = bi[j]; bi[j] = bi[j - 1]; bi[j - 1] = ti;
                }
            }
        }
    }

    // Merge: 10 rounds, each selects the global min of the 32 list heads.
    for (int r = 0; r < KSEL; ++r) {
        float v  = bv[0];
        int   id = bi[0];
        int   src = lane;
#pragma unroll
        for (int off = 16; off > 0; off >>= 1) {
            float ov  = __shfl_xor(v, off, 32);
            int   oid = __shfl_xor(id, off, 32);
            int   osr = __shfl_xor(src, off, 32);
            bool take = (ov < v) || (ov == v && oid < id);
            if (take) { v = ov; id = oid; src = osr; }
        }
        if (lane == 0) idx[qq * KSEL + r] = id;
        if (lane == src) {           // pop this lane's head
#pragma unroll
            for (int j = 0; j < KSEL - 1; ++j) { bv[j] = bv[j + 1]; bi[j] = bi[j + 1]; }
            bv[KSEL - 1] = 3.4e38f;
            bi[KSEL - 1] = 0x7fffffff;
        }
    }
}

// ---------------------------------------------------------------------------
// out[r][qq][...] = obs[idx[qq][r]][...]  -- the bandwidth-dominant step.
// One 256-thread block per output image; 128-bit non-temporal copy
// (streaming ~578 MB total; don't pollute L2, obs >> 192 MB L2).
// ---------------------------------------------------------------------------
__global__ void knn_gather(const float* __restrict__ obs,
                           const int* __restrict__ idx,
                           float* __restrict__ out) {
    const int t  = blockIdx.x;        // t = r*256 + qq
    const int r  = t >> 8;
    const int qq = t & 255;
    const int src = idx[qq * KSEL + r];

    const v4f* sp = (const v4f*)(obs + (size_t)src * CHW);
    v4f*       dp = (v4f*)(out + (size_t)t * CHW);
    for (int i = threadIdx.x; i < CHW4; i += 256) {
        v4f v = __builtin_nontemporal_load(sp + i);
        __builtin_nontemporal_store(v, dp + i);
    }
}

// ---------------------------------------------------------------------------
extern "C" void kernel_launch(void* const* d_in, const int* in_sizes, int n_in,
                              void* d_out, int out_size, void* d_ws, size_t ws_size,
                              hipStream_t stream) {
    const float* q   = (const float*)d_in[0];
    const float* k   = (const float*)d_in[1];
    const float* obs = (const float*)d_in[2];
    float* out = (float*)d_out;

    // Workspace: dist matrix (256*10000 f32 = 10.24 MB) | k2 (10016 f32) | idx (2560 i32)
    float* s  = (float*)d_ws;
    float* k2 = s + (size_t)QN * NN;
    int*  idx = (int*)(k2 + 10016);

    knn_k2_kernel<<<dim3((NN + 255) / 256), dim3(256), 0, stream>>>(k, k2);
    knn_dist_wmma<<<dim3(NN / 16, QN / 16), dim3(32), 0, stream>>>(q, k, k2, s);
    knn_topk<<<dim3(QN), dim3(32), 0, stream>>>(s, idx);
    knn_gather<<<dim3(QN * KSEL), dim3(256), 0, stream>>>(obs, idx, out);
}